// DifferentiablePreisachNNModel_8804682957422
// MI455X (gfx1250) — compile-verified
//
#include <hip/hip_runtime.h>
#include <hip/hip_bf16.h>
#include <cstddef>

typedef float v2f __attribute__((ext_vector_type(2)));
typedef float v8f __attribute__((ext_vector_type(8)));

#define HIDDEN  128
#define NLAYERS 3

// ---------------------------------------------------------------------------
// Kernel 1: density MLP.  One block computes 16 mesh rows.
//   8 waves x 16 cols = 128-wide hidden; 3 residual layers via
//   V_WMMA_F32_16X16X4_F32 (K-loop of 32 per layer), h tile staged in LDS.
// ---------------------------------------------------------------------------
__global__ __launch_bounds__(256) void density_kernel(
    const float* __restrict__ mesh, const float* __restrict__ w_in,
    const float* __restrict__ b_in, const float* __restrict__ wsL,
    const float* __restrict__ bs,   const float* __restrict__ w_out,
    const float* __restrict__ b_out, float* __restrict__ density, int n)
{
    __shared__ float hbuf[16][HIDDEN + 2];   // +2 pad: dodge 64-bank conflicts

    const int tid  = threadIdx.x;
    const int lane = tid & 31;
    const int wave = tid >> 5;               // 0..7 -> N tile
    const int row0 = blockIdx.x * 16;

    // ---- input layer: h = relu(mesh @ w_in + b_in) -> 16 x 128 tile ----
    for (int f = tid; f < 16 * HIDDEN; f += 256) {
        const int r = f >> 7, c = f & (HIDDEN - 1);
        const int row = row0 + r;
        float m0 = 0.f, m1 = 0.f;
        if (row < n) { m0 = mesh[2 * row + 0]; m1 = mesh[2 * row + 1]; }
        const float h = m0 * w_in[c] + m1 * w_in[HIDDEN + c] + b_in[c];
        hbuf[r][c] = h > 0.f ? h : 0.f;
    }
    __syncthreads();

    const int nIdx = (wave << 4) + (lane & 15);   // output column of this lane
    const int kHi  = (lane >> 4) << 1;            // +0 / +2 K sub-slot (A & B frags)
    const int mRow = lane & 15;                   // A-frag row
    const int mOff = (lane >> 4) << 3;            // C-frag M offset (0 or 8)

    for (int layer = 0; layer < NLAYERS; ++layer) {
        const float* W = wsL + (size_t)layer * HIDDEN * HIDDEN;
        v8f acc = {};
        #pragma unroll
        for (int k0 = 0; k0 < HIDDEN; k0 += 4) {
            const int ka = k0 + kHi;
            v2f a, b;
            a.x = hbuf[mRow][ka];
            a.y = hbuf[mRow][ka + 1];
            b.x = W[(size_t)ka       * HIDDEN + nIdx];
            b.y = W[(size_t)(ka + 1) * HIDDEN + nIdx];
            acc = __builtin_amdgcn_wmma_f32_16x16x4_f32(
                      false, a, false, b, (short)0, acc, false, false);
        }
        const float bias = bs[layer * HIDDEN + nIdx];
        __syncthreads();                     // all waves done reading hbuf
        #pragma unroll
        for (int v = 0; v < 8; ++v) {
            const float z = acc[v] + bias;
            const float r = z > 0.f ? z : 0.f;
            hbuf[mOff + v][nIdx] += r;       // residual: h += relu(hW + b)
        }
        __syncthreads();
    }

    // ---- output layer: density = sigmoid(h @ w_out + b_out) ----
    if (tid < 16) {
        const int row = row0 + tid;
        if (row < n) {
            float z = b_out[0];
            #pragma unroll 8
            for (int c = 0; c < HIDDEN; ++c) z += hbuf[tid][c] * w_out[c];
            density[row] = 1.0f / (1.0f + __expf(-z));
        }
    }
}

// ---------------------------------------------------------------------------
// Kernel 2: deterministic reduction of density -> den_sum (d_ws[0])
// ---------------------------------------------------------------------------
__global__ __launch_bounds__(256) void densum_kernel(
    const float* __restrict__ density, float* __restrict__ den_sum, int n)
{
    __shared__ float red[256];
    float s = 0.f;
    for (int i = threadIdx.x; i < n; i += 256) s += density[i];
    red[threadIdx.x] = s;
    __syncthreads();
    for (int off = 128; off > 0; off >>= 1) {
        if (threadIdx.x < off) red[threadIdx.x] += red[threadIdx.x + off];
        __syncthreads();
    }
    if (threadIdx.x == 0) den_sum[0] = red[0];
}

// ---------------------------------------------------------------------------
// Kernel 3: Preisach scan.  One thread = one (batch, mesh-point) chain,
// sequential over T=2048 steps.  x row staged in LDS; stores are the only
// global traffic in the loop (coalesced over contiguous j).
// ---------------------------------------------------------------------------
__global__ __launch_bounds__(64) void scan_kernel(
    const float* __restrict__ x, const float* __restrict__ mesh,
    const float* __restrict__ init_raw, float* __restrict__ states,
    int n, int T)
{
    __shared__ float xs[2048];
    const int b = blockIdx.y;
    const int j = blockIdx.x * 64 + threadIdx.x;

    for (int t = threadIdx.x; t < T; t += 64) xs[t] = x[(size_t)b * T + t];
    __syncthreads();

    float beta = 0.f, alpha = 0.f, s = 0.f;
    const bool act = (j < n);
    if (act) {
        beta  = mesh[2 * j + 0];
        alpha = mesh[2 * j + 1];
        const float v = init_raw[j];
        s = (v > 0.f) ? 1.f : (v < 0.f ? -1.f : 0.f);
    }

    float* out = states + (size_t)b * T * n + j;
    const float invT = 1000.0f;              // 1 / TEMP
    for (int t = 0; t < T; ++t) {
        const float u  = xs[t];
        const float up = 1.f / (1.f + __expf(-(u - alpha) * invT));
        const float dn = 1.f / (1.f + __expf(-(beta - u) * invT));
        s = up + (1.f - up) * (-dn + (1.f - dn) * s);
        if (act) out[(size_t)t * n] = s;
    }
}

// ---------------------------------------------------------------------------
// Kernel 4: m[b,t] = m_scale * (states[b,t,:].density / den_sum) + m_offset.
// One wave per (b,t) row; density cached in LDS; wave32 shfl_xor reduce.
// ---------------------------------------------------------------------------
__global__ __launch_bounds__(256) void m_kernel(
    const float* __restrict__ states, const float* __restrict__ density,
    const float* __restrict__ den_sum, const float* __restrict__ raw_scale,
    const float* __restrict__ raw_off, float* __restrict__ m_out,
    int n, int total)
{
    __shared__ float dsh[5152];
    for (int i = threadIdx.x; i < n; i += 256) dsh[i] = density[i];
    __syncthreads();

    const int lane = threadIdx.x & 31;
    const int wave = threadIdx.x >> 5;
    const int row  = blockIdx.x * 8 + wave;
    if (row >= total) return;

    const float* sp = states + (size_t)row * n;
    float acc = 0.f;
    for (int i = lane; i < n; i += 32) acc += sp[i] * dsh[i];
    #pragma unroll
    for (int off = 16; off > 0; off >>= 1) acc += __shfl_xor(acc, off, 32);

    if (lane == 0) {
        const float m_scale  = 10.0f * (1.f / (1.f + __expf(-raw_scale[0])));
        const float m_offset = -10.0f + 20.0f * (1.f / (1.f + __expf(-raw_off[0])));
        m_out[row] = m_scale * (acc / den_sum[0]) + m_offset;
    }
}

// ---------------------------------------------------------------------------
extern "C" void kernel_launch(void* const* d_in, const int* in_sizes, int n_in,
                              void* d_out, int out_size, void* d_ws, size_t ws_size,
                              hipStream_t stream)
{
    const float* x        = (const float*)d_in[0];
    const float* mesh     = (const float*)d_in[1];
    const float* w_in     = (const float*)d_in[2];
    const float* b_in     = (const float*)d_in[3];
    const float* wsL      = (const float*)d_in[4];
    const float* bs       = (const float*)d_in[5];
    const float* w_out    = (const float*)d_in[6];
    const float* b_out    = (const float*)d_in[7];
    const float* raw_s    = (const float*)d_in[8];
    const float* raw_o    = (const float*)d_in[9];
    const float* init_raw = (const float*)d_in[10];

    const int n = in_sizes[10];          // 5151 mesh points
    const int B = 4;
    const int T = in_sizes[0] / B;       // 2048

    float* m_out       = (float*)d_out;                       // [B*T]
    float* density_out = m_out + (size_t)B * T;               // [n]
    float* states_out  = density_out + n;                     // [B*T*n]
    float* den_sum     = (float*)d_ws;                        // [1]

    const int rowTiles = (n + 15) / 16;
    density_kernel<<<rowTiles, 256, 0, stream>>>(mesh, w_in, b_in, wsL, bs,
                                                 w_out, b_out, density_out, n);
    densum_kernel<<<1, 256, 0, stream>>>(density_out, den_sum, n);

    dim3 sg((n + 63) / 64, B);
    scan_kernel<<<sg, 64, 0, stream>>>(x, mesh, init_raw, states_out, n, T);

    const int total = B * T;
    m_kernel<<<(total + 7) / 8, 256, 0, stream>>>(states_out, density_out,
                                                  den_sum, raw_s, raw_o,
                                                  m_out, n, total);
}